// RetinaDecoder_44676249813200
// MI455X (gfx1250) — compile-verified
//
#include <hip/hip_runtime.h>
#include <hip/hip_bf16.h>
#include <cstdint>

#define NLVL 5
#define BATCH 8
#define NCLS 80
#define MSEL 4125        // 1000+1000+1000+900+225 selected per image
#define MAXDET 100
#define NEGV -1.0e9f

// ---------------------------------------------------------------------------
// CDNA5 async global -> LDS copy (ASYNCcnt path, CDNA5 ISA)
// vdst = LDS byte offset (relative to wave LDS base), vaddr = 64-bit global.
// ---------------------------------------------------------------------------
__device__ __forceinline__ void async_copy_b128(unsigned lds_off, const void* g) {
    unsigned long long ga = (unsigned long long)(uintptr_t)g;
    asm volatile("global_load_async_to_lds_b128 %0, %1, off"
                 :: "v"(lds_off), "v"(ga) : "memory");
}
__device__ __forceinline__ void wait_async0() {
    asm volatile("s_wait_asynccnt 0" ::: "memory");
}

// ---------------------------------------------------------------------------
// K1: per-anchor max/argmax over 80 classes.  Each block stages 128 rows
// (40 KB) into LDS via async B128 loads (coalesced 16B/lane), then each lane
// reduces one row.  Bandwidth-bound: 196 MB total -> ~9us at 23.3 TB/s.
// ---------------------------------------------------------------------------
__global__ __launch_bounds__(128) void k_score_argmax(
    const float* __restrict__ cls, float* __restrict__ scores,
    int* __restrict__ classes, int total) {
    __shared__ float smem[128 * NCLS];                    // 40 KB, LDS offset 0
    const int tid = threadIdx.x;
    const long long base = (long long)blockIdx.x * 128;
    int rows = total - (int)base; if (rows > 128) rows = 128;
    const int nvec = rows * (NCLS / 4);                   // float4 chunks
    const float* gbase = cls + base * NCLS;
    for (int i = tid; i < nvec; i += 128)
        async_copy_b128((unsigned)(i * 16), (const void*)(gbase + i * 4));
    wait_async0();
    __syncthreads();
    if (tid < rows) {
        const float* row = &smem[tid * NCLS];
        float best = row[0]; int bc = 0;
        #pragma unroll
        for (int c = 1; c < NCLS; ++c) {
            float v = row[c];
            if (v > best) { best = v; bc = c; }           // first-index argmax
        }
        scores[base + tid]  = best;
        classes[base + tid] = bc;
    }
}

// ---------------------------------------------------------------------------
// K2: K-th largest score per (level,batch) via binary search on the bit
// pattern (scores are positive -> uint order == float order).  One block per
// batch image.  Invariant: count(>=lo) >= K, count(>=hi) < K.
// ---------------------------------------------------------------------------
__global__ __launch_bounds__(256) void k_kth_threshold(
    const float* __restrict__ scores, unsigned* __restrict__ thr, int N, int K) {
    const unsigned* s = (const unsigned*)scores + (long long)blockIdx.x * N;
    __shared__ int cnt_sh;
    unsigned lo = 0u, hi = 0x7F800000u;
    while (lo + 1u < hi) {
        unsigned mid = lo + ((hi - lo) >> 1);
        if (threadIdx.x == 0) cnt_sh = 0;
        __syncthreads();
        int c = 0;
        for (int j = threadIdx.x; j < N; j += 256) c += (s[j] >= mid) ? 1 : 0;
        atomicAdd(&cnt_sh, c);
        __syncthreads();
        int tot = cnt_sh;
        __syncthreads();                                  // all read before reset
        if (tot >= K) lo = mid; else hi = mid;
    }
    if (threadIdx.x == 0) thr[blockIdx.x] = lo;
}

// ---------------------------------------------------------------------------
// K3a: init selected-score slots to NEG and zero the compaction counters.
// ---------------------------------------------------------------------------
__global__ void k_init(float* sel_s, int* counters) {
    int i = blockIdx.x * blockDim.x + threadIdx.x;
    if (i < BATCH * MSEL) sel_s[i] = NEGV;
    if (i < NLVL * BATCH) counters[i] = 0;
}

// ---------------------------------------------------------------------------
// K3b: compact survivors (>= threshold, capped at K) and decode boxes with
// the reference's int-truncation + one-sided clamps.
// ---------------------------------------------------------------------------
__global__ __launch_bounds__(256) void k_select_decode(
    const float* __restrict__ scores, const int* __restrict__ classes,
    const float* __restrict__ reg, const float* __restrict__ anc,
    const unsigned* __restrict__ thr, int* __restrict__ counters,
    float* __restrict__ sel_s, float* __restrict__ sel_c, float* __restrict__ sel_b,
    int N, int K, int selOff, int useThr) {
    long long g = (long long)blockIdx.x * 256 + threadIdx.x;
    if (g >= (long long)BATCH * N) return;
    int b = (int)(g / N);
    int a = (int)(g % N);
    float sc = scores[g];
    int slot;
    if (useThr) {
        if (__float_as_uint(sc) < thr[b]) return;
        slot = atomicAdd(&counters[b], 1);
        if (slot >= K) return;
    } else {
        slot = a;                                         // pass-through level
    }
    const float* rg = reg + g * 4;
    const float* an = anc + g * 4;
    float w  = an[2] - an[0], h  = an[3] - an[1];
    float cx = an[0] + 0.5f * w, cy = an[1] + 0.5f * h;
    float pw  = expf(rg[2] * 0.2f) * w;
    float ph  = expf(rg[3] * 0.2f) * h;
    float pcx = rg[0] * 0.1f * w + cx;
    float pcy = rg[1] * 0.1f * h + cy;
    int x1 = (int)(pcx - 0.5f * pw);                      // trunc toward zero
    int y1 = (int)(pcy - 0.5f * ph);
    int x2 = (int)(pcx + 0.5f * pw);
    int y2 = (int)(pcy + 0.5f * ph);
    x1 = x1 > 0 ? x1 : 0;
    y1 = y1 > 0 ? y1 : 0;
    x2 = x2 < 639 ? x2 : 639;
    y2 = y2 < 639 ? y2 : 639;
    long long o = (long long)b * MSEL + selOff + slot;
    sel_s[o] = sc;
    sel_c[o] = (float)classes[g];
    sel_b[o * 4 + 0] = (float)x1;
    sel_b[o * 4 + 1] = (float)y1;
    sel_b[o * 4 + 2] = (float)x2;
    sel_b[o * 4 + 3] = (float)y2;
}

// ---------------------------------------------------------------------------
// K4: greedy NMS.  One block (32 wave32 waves) per image; `live` and `area`
// in LDS (33 KB).  100 serial rounds of wave-shuffle argmax + suppression.
// ---------------------------------------------------------------------------
__global__ __launch_bounds__(1024) void k_nms(
    const float* __restrict__ sel_s, const float* __restrict__ sel_c,
    const float* __restrict__ sel_b, float* __restrict__ out) {
    __shared__ float live[MSEL];
    __shared__ float area[MSEL];
    __shared__ float rv[32];
    __shared__ int   ri[32];
    __shared__ float tb[4];
    __shared__ int   s_fi, s_has;
    const int b = blockIdx.x, tid = threadIdx.x;
    const float* ss  = sel_s + (long long)b * MSEL;
    const float* bbx = sel_b + (long long)b * MSEL * 4;
    for (int j = tid; j < MSEL; j += 1024) {
        float s = ss[j];
        live[j] = (s > 0.05f) ? s : NEGV;                 // MIN_SCORE mask
        float x1 = bbx[j*4+0], y1 = bbx[j*4+1], x2 = bbx[j*4+2], y2 = bbx[j*4+3];
        area[j] = (x2 - x1) * (y2 - y1);
    }
    __syncthreads();
    for (int it = 0; it < MAXDET; ++it) {
        // ---- argmax(live), first-index tie-break ----
        float bv = -3.0e38f; int bi = 0x7FFFFFFF;
        for (int j = tid; j < MSEL; j += 1024) {
            float v = live[j];
            if (v > bv) { bv = v; bi = j; }
        }
        for (int off = 16; off > 0; off >>= 1) {          // wave32 reduce
            float ov = __shfl_down(bv, off);
            int   oi = __shfl_down(bi, off);
            if (ov > bv || (ov == bv && oi < bi)) { bv = ov; bi = oi; }
        }
        if ((tid & 31) == 0) { rv[tid >> 5] = bv; ri[tid >> 5] = bi; }
        __syncthreads();
        if (tid == 0) {
            float fv = rv[0]; int fi = ri[0];
            for (int k = 1; k < 32; ++k) {
                float v = rv[k]; int i2 = ri[k];
                if (v > fv || (v == fv && i2 < fi)) { fv = v; fi = i2; }
            }
            int has = fv > (NEGV * 0.5f);
            s_fi = fi; s_has = has;
            float bx1 = bbx[fi*4+0], by1 = bbx[fi*4+1];
            float bx2 = bbx[fi*4+2], by2 = bbx[fi*4+3];
            tb[0] = bx1; tb[1] = by1; tb[2] = bx2; tb[3] = by2;
            out[b*MAXDET + it]                = has ? ss[fi] : -1.0f;
            out[BATCH*MAXDET + b*MAXDET + it] = has ? sel_c[(long long)b*MSEL + fi] : -1.0f;
            long long ob = 2LL*BATCH*MAXDET + ((long long)b*MAXDET + it)*4;
            out[ob+0] = has ? bx1 : -1.0f;
            out[ob+1] = has ? by1 : -1.0f;
            out[ob+2] = has ? bx2 : -1.0f;
            out[ob+3] = has ? by2 : -1.0f;
            live[fi] = NEGV;                              // remove selected box
        }
        __syncthreads();
        if (s_has) {
            const int fi = s_fi;
            const float tx1 = tb[0], ty1 = tb[1], tx2 = tb[2], ty2 = tb[3];
            const float ta = area[fi];
            for (int j = tid; j < MSEL; j += 1024) {
                float ix1 = fmaxf(bbx[j*4+0], tx1);
                float iy1 = fmaxf(bbx[j*4+1], ty1);
                float ix2 = fminf(bbx[j*4+2], tx2);
                float iy2 = fminf(bbx[j*4+3], ty2);
                float iw = fmaxf(ix2 - ix1, 0.0f);
                float ih = fmaxf(iy2 - iy1, 0.0f);
                float inter = iw * ih;
                float uni = fmaxf(ta + area[j] - inter, 1e-4f);
                if (inter / uni >= 0.5f) live[j] = NEGV;
            }
        }
        __syncthreads();
    }
}

// ---------------------------------------------------------------------------
// Launcher.  Inputs per level i: d_in[3i]=cls, d_in[3i+1]=reg, d_in[3i+2]=anc.
// Output: [scores 8x100 | classes 8x100 | boxes 8x100x4] flat float32.
// ---------------------------------------------------------------------------
extern "C" void kernel_launch(void* const* d_in, const int* in_sizes, int n_in,
                              void* d_out, int out_size, void* d_ws, size_t ws_size,
                              hipStream_t stream) {
    static const int Ns[NLVL]   = {57600, 14400, 3600, 900, 225};
    static const int Ks[NLVL]   = {1000, 1000, 1000, 900, 225};
    static const int sOff[NLVL] = {0, 1000, 2000, 3000, 3900};
    static const long long lvlOff[NLVL] = {0, 460800, 576000, 604800, 612000};
    const long long SC_TOT = 613800;     // sum_i B*N_i

    char* ws = (char*)d_ws;
    float*    scoresW  = (float*)ws;                               // 613800 f
    int*      classesW = (int*)(ws + SC_TOT * 4);                  // 613800 i
    unsigned* thrW     = (unsigned*)(ws + SC_TOT * 8);             // 40 u32
    int*      cntW     = (int*)(ws + SC_TOT * 8 + 256);            // 40 i32
    float*    selS     = (float*)(ws + SC_TOT * 8 + 512);          // 33000 f
    float*    selC     = selS + (long long)BATCH * MSEL;           // 33000 f
    float*    selB     = selC + (long long)BATCH * MSEL;           // 132000 f
    float*    out      = (float*)d_out;

    k_init<<<(BATCH * MSEL + 255) / 256, 256, 0, stream>>>(selS, cntW);

    for (int l = 0; l < NLVL; ++l) {
        int total = BATCH * Ns[l];
        k_score_argmax<<<(total + 127) / 128, 128, 0, stream>>>(
            (const float*)d_in[3 * l], scoresW + lvlOff[l], classesW + lvlOff[l], total);
    }
    for (int l = 0; l < 3; ++l) {                         // levels with N >= TOP_N
        k_kth_threshold<<<BATCH, 256, 0, stream>>>(
            scoresW + lvlOff[l], thrW + l * BATCH, Ns[l], Ks[l]);
    }
    for (int l = 0; l < NLVL; ++l) {
        int total = BATCH * Ns[l];
        int useThr = (Ns[l] >= 1000) ? 1 : 0;
        k_select_decode<<<(total + 255) / 256, 256, 0, stream>>>(
            scoresW + lvlOff[l], classesW + lvlOff[l],
            (const float*)d_in[3 * l + 1], (const float*)d_in[3 * l + 2],
            thrW + l * BATCH, cntW + l * BATCH,
            selS, selC, selB, Ns[l], Ks[l], sOff[l], useThr);
    }
    k_nms<<<BATCH, 1024, 0, stream>>>(selS, selC, selB, out);
}